// FuzzyMultiLayer_10857677324319
// MI455X (gfx1250) — compile-verified
//
#include <hip/hip_runtime.h>
#include <hip/hip_bf16.h>
#include <math.h>

typedef __attribute__((ext_vector_type(2))) float v2f;
typedef __attribute__((ext_vector_type(8))) float v8f;

// -------------------------------------------------------------------------
// Prep kernel: invert the 8 lower-triangular 4x4 scale matrices, build
//   ws[0..127]   : M[32][4]  row-major, M[4k+i][c] = Linv_k[i][c]
//   ws[128..159] : bias[32], bias[4k+i] = -sum_c Linv_k[i][c]*mu_k[c]
//   ws[160..167] : lc[8],    lc[k] = -2*log(2*pi) - sum_i log|L_k[i][i]|
// Trivial work; one thread does it all.
// -------------------------------------------------------------------------
__global__ void fuzzy_prep_kernel(const float* __restrict__ mean,
                                  const float* __restrict__ scale,
                                  float* __restrict__ ws) {
  if (threadIdx.x != 0 || blockIdx.x != 0) return;
  for (int k = 0; k < 8; ++k) {
    float L[4][4];
    for (int i = 0; i < 4; ++i)
      for (int j = 0; j < 4; ++j)
        L[i][j] = (j <= i) ? scale[(k * 4 + i) * 4 + j] : 0.0f;
    float Inv[4][4] = {};
    for (int i = 0; i < 4; ++i) {
      Inv[i][i] = 1.0f / L[i][i];
      for (int j = 0; j < i; ++j) {
        float s = 0.0f;
        for (int m = j; m < i; ++m) s += L[i][m] * Inv[m][j];
        Inv[i][j] = -s / L[i][i];
      }
    }
    float logdet = 0.0f;
    for (int i = 0; i < 4; ++i) logdet += logf(fabsf(L[i][i]));
    for (int i = 0; i < 4; ++i) {
      const int comp = k * 4 + i;
      float bsum = 0.0f;
      for (int c = 0; c < 4; ++c) {
        ws[comp * 4 + c] = Inv[i][c];
        bsum += Inv[i][c] * mean[k * 4 + c];
      }
      ws[128 + comp] = -bsum;
    }
    // -0.5*C*log(2*pi) with C=4
    ws[160 + k] = -3.6757541328186907f - logdet;
  }
}

// -------------------------------------------------------------------------
// Main kernel. One wave handles a group of 32 consecutive pixels.
//
// WMMA orientation: D[comp][pixel] = sum_c M[comp][c] * x[c][pixel] + bias.
//   A (16x4 f32): lane L<16 -> comp L, ch {0,1}; lane L+16 -> comp L, ch {2,3}
//   B (4x16 f32): vgpr0 lanes0-15=(K=0,N=pix), lanes16-31=(K=2,N=pix); vgpr1 K=1/3
//   D: vgpr r, lanes0-15 -> comp r, pixel=lane; lanes16-31 -> comp r+8, pixel=lane-16
// So per lane, one pixel's components live in the 8 accumulator VGPRs:
// class sums of z^2 are in-lane adds. Two A matrices (comps 0-15, 16-31).
// -------------------------------------------------------------------------
__launch_bounds__(256)
__global__ void fuzzy_main_kernel(const float* __restrict__ x,
                                  const float* __restrict__ ws,
                                  float* __restrict__ out,
                                  int ngroups) {
  constexpr int PSTR = 9;                 // padded class stride (bank-friendly)
  constexpr int WLDS = 32 * PSTR + 32 * 4;  // p[32][9] + x[32][4] = 416 floats
  __shared__ float lds[8 * WLDS];
  const int lane = threadIdx.x & 31;
  const int wave = threadIdx.x >> 5;
  float* p_lds = lds + wave * WLDS;
  float* x_lds = p_lds + 32 * PSTR;

  const float* Mw = ws;         // [32][4]
  const float* bw = ws + 128;   // [32]
  const float* lw = ws + 160;   // [8]

  const bool lo   = (lane < 16);
  const int  ch0  = lo ? 0 : 2;       // channels this lane supplies to A/B
  const int  pix  = lane & 15;        // pixel-in-subtile (also A row index)
  const int  half8 = lo ? 0 : 8;      // D comps offset for this lane half
  const int  cb   = lo ? 0 : 2;       // class base held by this lane

  // A operands (constant): comps 0-15 and 16-31 of M
  const v2f A0 = { Mw[pix * 4 + ch0],        Mw[pix * 4 + ch0 + 1] };
  const v2f A1 = { Mw[(pix + 16) * 4 + ch0], Mw[(pix + 16) * 4 + ch0 + 1] };
  // C operands: bias broadcast along the pixel (N) dimension
  v8f c0base, c1base;
#pragma unroll
  for (int r = 0; r < 8; ++r) {
    c0base[r] = bw[half8 + r];
    c1base[r] = bw[16 + half8 + r];
  }
  const float lcA0 = lw[cb],     lcA1 = lw[cb + 1];
  const float lcB0 = lw[cb + 4], lcB1 = lw[cb + 5];

  const int k_out = lane >> 2;   // output phase: lane == (k,c) pair
  const int c_out = lane & 3;

  const int nw = (int)(gridDim.x * (blockDim.x >> 5));
  int grp = (int)(blockIdx.x * (blockDim.x >> 5) + wave);
  for (; grp < ngroups; grp += nw) {
    const float* xg = x + (size_t)grp * 128;   // 32 pixels * 4 channels
    __builtin_prefetch(xg + (size_t)nw * 128, 0, 1);

#pragma unroll
    for (int sub = 0; sub < 2; ++sub) {
      // B operand: x tile, channels x 16 pixels. Also staged to LDS.
      const float2 xv = *(const float2*)(xg + sub * 64 + pix * 4 + ch0);
      *(float2*)(x_lds + (pix + sub * 16) * 4 + ch0) = xv;
      const v2f Bx = { xv.x, xv.y };

      v8f z0 = __builtin_amdgcn_wmma_f32_16x16x4_f32(
          false, A0, false, Bx, (short)0, c0base, false, false);
      v8f z1 = __builtin_amdgcn_wmma_f32_16x16x4_f32(
          false, A1, false, Bx, (short)0, c1base, false, false);

      // Per-class squared Mahalanobis (all in-lane: comps are in VGPRs)
      const float qa = z0[0]*z0[0] + z0[1]*z0[1] + z0[2]*z0[2] + z0[3]*z0[3];
      const float qb = z0[4]*z0[4] + z0[5]*z0[5] + z0[6]*z0[6] + z0[7]*z0[7];
      const float qc = z1[0]*z1[0] + z1[1]*z1[1] + z1[2]*z1[2] + z1[3]*z1[3];
      const float qd = z1[4]*z1[4] + z1[5]*z1[5] + z1[6]*z1[6] + z1[7]*z1[7];
      const float pa = __expf(lcA0 - 0.5f * qa);   // class cb
      const float pb = __expf(lcA1 - 0.5f * qb);   // class cb+1
      const float pc = __expf(lcB0 - 0.5f * qc);   // class cb+4
      const float pd = __expf(lcB1 - 0.5f * qd);   // class cb+5

      const int prow = (pix + sub * 16) * PSTR;
      p_lds[prow + cb]     = pa;
      p_lds[prow + cb + 1] = pb;
      p_lds[prow + cb + 4] = pc;
      p_lds[prow + cb + 5] = pd;
    }

    // Same-wave LDS ordering: wait for DS writes before the pivot reads.
    asm volatile("s_wait_dscnt 0" ::: "memory");

    // Output phase: lane = (k,c) plane; normalize over the 32-pixel group
    // (matches the reference's post-reshape l2_normalize axis).
    float g[32];
    float s = 0.0f;
#pragma unroll
    for (int j = 0; j < 32; ++j) {
      const float gv = p_lds[j * PSTR + k_out] * x_lds[j * 4 + c_out];
      g[j] = gv;
      s += gv * gv;
    }
    const float inv = rsqrtf(fmaxf(s, 1e-12f));
    // out flat = kc*2^21 + b*2^16 + hw = kc*2^21 + global_pixel
    float* op = out + ((size_t)lane << 21) + (size_t)grp * 32;
#pragma unroll
    for (int j4 = 0; j4 < 8; ++j4) {
      float4 o = { g[j4 * 4 + 0] * inv, g[j4 * 4 + 1] * inv,
                   g[j4 * 4 + 2] * inv, g[j4 * 4 + 3] * inv };
      *(float4*)(op + j4 * 4) = o;   // each lane fills exactly one 128B line
    }
    asm volatile("" ::: "memory");
  }
}

extern "C" void kernel_launch(void* const* d_in, const int* in_sizes, int n_in,
                              void* d_out, int out_size, void* d_ws, size_t ws_size,
                              hipStream_t stream) {
  const float* x     = (const float*)d_in[0];   // [32,256,256,4] f32
  const float* mean  = (const float*)d_in[1];   // [8,4] f32
  const float* scale = (const float*)d_in[2];   // [8,4,4] f32
  float* out = (float*)d_out;                   // 67108864 f32
  float* ws  = (float*)d_ws;                    // needs 168 floats

  fuzzy_prep_kernel<<<1, 64, 0, stream>>>(mean, scale, ws);

  const int npix    = in_sizes[0] / 4;   // B*H*W = 2097152
  const int ngroups = npix / 32;         // 65536
  fuzzy_main_kernel<<<2048, 256, 0, stream>>>(x, ws, out, ngroups);
}